// RoIBBox_74122545594378
// MI455X (gfx1250) — compile-verified
//
#include <hip/hip_runtime.h>
#include <cstdint>

#define BATCH 8
#define FH 32
#define FW 256
#define AA 6
#define NANCH (FH*FW*AA)        // 49152
#define PRE_TOPN 2000
#define KP 2048                 // padded (pow2) for bitonic sort
#define POST_TOPN 300
#define NMS_IOU_TH 0.7f
#define MASKW 64                // mask words per row (63 used, padded for alignment)
#define NBLK 63                 // ceil(2000/32) column blocks

typedef __attribute__((ext_vector_type(2))) float v2f;
typedef __attribute__((ext_vector_type(8))) float v8f;

// ---- monotone float <-> uint mapping for radix select / sorting ----
__device__ __forceinline__ unsigned flip_f32(float f) {
  unsigned u = __float_as_uint(f);
  return u ^ ((u & 0x80000000u) ? 0xFFFFFFFFu : 0x80000000u);
}
__device__ __forceinline__ float unflip_f32(unsigned u) {
  u ^= ((u & 0x80000000u) ? 0x80000000u : 0xFFFFFFFFu);
  return __uint_as_float(u);
}

// =====================================================================
// Kernel 1: softmax over A=6 logits per location + box decode
// =====================================================================
__global__ void decode_softmax_kernel(const float* __restrict__ deltas,
                                      const float* __restrict__ logits,
                                      const float* __restrict__ anchors,
                                      float* __restrict__ scoresAll,
                                      float* __restrict__ boxesAll) {
  int gid = blockIdx.x * blockDim.x + threadIdx.x;
  if (gid >= BATCH * NANCH) return;
  int b = gid / NANCH;
  int t = gid - b * NANCH;
  int loc = t / AA;
  int a = t - loc * AA;

  const float* lg = logits + (size_t)b * NANCH + (size_t)loc * AA;
  float m = lg[0];
  #pragma unroll
  for (int k = 1; k < AA; ++k) m = fmaxf(m, lg[k]);
  float s = 0.f;
  #pragma unroll
  for (int k = 0; k < AA; ++k) s += expf(lg[k] - m);
  float score = expf(lg[a] - m) / s;

  const float* dp = deltas + (size_t)gid * 4;
  float d0 = dp[0] * 0.1f, d1 = dp[1] * 0.1f;
  float d2 = dp[2] * 0.2f, d3 = dp[3] * 0.2f;
  const float* ap = anchors + (size_t)t * 4;
  float ah = ap[2] - ap[0], aw = ap[3] - ap[1];
  float acy = ap[0] + 0.5f * ah, acx = ap[1] + 0.5f * aw;
  float bh = expf(d2) * ah, bw = expf(d3) * aw;
  float bcy = d0 * ah + acy, bcx = d1 * aw + acx;
  float y1 = bcy - 0.5f * bh, x1 = bcx - 0.5f * bw;

  float* bp = boxesAll + (size_t)gid * 4;
  bp[0] = y1; bp[1] = x1; bp[2] = y1 + bh; bp[3] = x1 + bw;
  scoresAll[gid] = score;
}

// =====================================================================
// Kernel 2: per-batch exact top-2000 via 4-round radix select, then
// LDS bitonic sort (descending score, ascending index tiebreak).
// One 1024-thread workgroup per batch; 2048x8B keys live in LDS.
// =====================================================================
__global__ __launch_bounds__(1024)
void select_sort_kernel(const float* __restrict__ scoresAll,
                        const float* __restrict__ boxesAll,
                        float* __restrict__ selScores,
                        float* __restrict__ selBoxes,
                        float* __restrict__ selArea) {
  __shared__ unsigned hist[256];
  __shared__ int sh_byte, sh_k;
  __shared__ unsigned cntG, cntE;
  __shared__ unsigned long long skey[KP];

  const int b = blockIdx.x;
  const int tid = threadIdx.x;
  const float* sc = scoresAll + (size_t)b * NANCH;

  unsigned pfx = 0u, pmask = 0u;
  if (tid == 0) sh_k = PRE_TOPN;
  __syncthreads();

  for (int round = 3; round >= 0; --round) {
    for (int j = tid; j < 256; j += 1024) hist[j] = 0u;
    __syncthreads();
    for (int i = tid; i < NANCH; i += 1024) {
      unsigned u = flip_f32(sc[i]);
      if ((u & pmask) == pfx) atomicAdd(&hist[(u >> (8 * round)) & 255u], 1u);
    }
    __syncthreads();
    if (tid == 0) {
      int k = sh_k;
      int bin = 255;
      for (; bin > 0; --bin) {
        if (k <= (int)hist[bin]) break;
        k -= (int)hist[bin];
      }
      sh_byte = bin;
      sh_k = k;
    }
    __syncthreads();
    pfx   |= ((unsigned)sh_byte) << (8 * round);
    pmask |= 0xFFu << (8 * round);
    __syncthreads();
  }

  const unsigned T = pfx;          // key of the 2000th-largest score
  const int tieTake = sh_k;        // how many equal-to-T keys to accept
  const int nGreater = PRE_TOPN - tieTake;

  if (tid == 0) { cntG = 0u; cntE = 0u; }
  for (int i = tid; i < KP; i += 1024) skey[i] = 0ull;
  __syncthreads();

  for (int i = tid; i < NANCH; i += 1024) {
    unsigned u = flip_f32(sc[i]);
    if (u > T) {
      unsigned slot = atomicAdd(&cntG, 1u);
      skey[slot] = ((unsigned long long)u << 32) | (unsigned)(~(unsigned)i);
    } else if (u == T) {
      unsigned e = atomicAdd(&cntE, 1u);
      if ((int)e < tieTake)
        skey[nGreater + e] = ((unsigned long long)u << 32) | (unsigned)(~(unsigned)i);
    }
  }
  __syncthreads();

  // bitonic sort, descending (composite key: score desc, index asc)
  for (unsigned size = 2; size <= KP; size <<= 1) {
    for (unsigned stride = size >> 1; stride > 0; stride >>= 1) {
      for (unsigned i = tid; i < KP; i += 1024) {
        unsigned j = i ^ stride;
        if (j > i) {
          bool desc = ((i & size) == 0u);
          unsigned long long x = skey[i], y = skey[j];
          if (desc ? (x < y) : (x > y)) { skey[i] = y; skey[j] = x; }
        }
      }
      __syncthreads();
    }
  }

  // emit sorted boxes/scores/areas; pad [2000,2048) with inert boxes
  for (int i = tid; i < KP; i += 1024) {
    float* ob = selBoxes + ((size_t)b * KP + i) * 4;
    if (i < PRE_TOPN) {
      unsigned long long key = skey[i];
      unsigned idx = ~(unsigned)(key & 0xFFFFFFFFull);
      float s = unflip_f32((unsigned)(key >> 32));
      const float* ib = boxesAll + ((size_t)b * NANCH + idx) * 4;
      float y1 = ib[0], x1 = ib[1], y2 = ib[2], x2 = ib[3];
      ob[0] = y1; ob[1] = x1; ob[2] = y2; ob[3] = x2;
      selScores[(size_t)b * KP + i] = s;
      selArea[(size_t)b * KP + i] = (y2 - y1) * (x2 - x1);
    } else {
      ob[0] = 1e9f; ob[1] = 1e9f; ob[2] = 1e9f; ob[3] = 1e9f;  // zero-area point far away
      selScores[(size_t)b * KP + i] = -1.0f;
      selArea[(size_t)b * KP + i] = 0.f;
    }
  }
}

// =====================================================================
// Kernel 3: NMS suppression bitmask. One wave32 per 16-row x 32-col tile.
// union = area[i] + area[j] - inter; the area outer-sum for each 16x16
// subtile is computed on the matrix pipe via V_WMMA_F32_16X16X4_F32
// (A = [area_row, 1] 16x4, B = [1 ; area_col] 4x16 -> D = outer sum),
// co-executing with the VALU min/max intersection chain.
// The IoU threshold test is division-free:
//   inter/max(uni,eps) > TH  <=>  inter > TH*max(uni,eps)
// (both sides non-negative), avoiding the v_div_scale/v_rcp fixup chain.
// Suppression bits are gathered wave-wide with ballot_w32.
// =====================================================================
__global__ __launch_bounds__(32)
void nms_mask_kernel(const float* __restrict__ selBoxes,
                     const float* __restrict__ selArea,
                     unsigned* __restrict__ mask) {
  __shared__ float cbx[32][4];
  __shared__ float car[32];
  __shared__ float rbx[16][4];
  __shared__ float rar[16];
  __shared__ unsigned wstage[16];

  const int cb = blockIdx.x;      // column block 0..62 (32 cols each)
  const int rt = blockIdx.y;      // row tile 0..124 (16 rows each)
  const int b  = blockIdx.z;
  const int lane = threadIdx.x;   // 0..31
  const int rowBase = rt * 16;
  const int colBase = cb * 32;

  {
    const float4 cv = *(const float4*)(selBoxes + ((size_t)b * KP + colBase + lane) * 4);
    cbx[lane][0] = cv.x; cbx[lane][1] = cv.y; cbx[lane][2] = cv.z; cbx[lane][3] = cv.w;
    car[lane] = selArea[(size_t)b * KP + colBase + lane];
    if (lane < 16) {
      const float4 rv = *(const float4*)(selBoxes + ((size_t)b * KP + rowBase + lane) * 4);
      rbx[lane][0] = rv.x; rbx[lane][1] = rv.y; rbx[lane][2] = rv.z; rbx[lane][3] = rv.w;
      rar[lane] = selArea[(size_t)b * KP + rowBase + lane];
    }
  }
  __syncthreads();

  // A matrix 16x4 f32 (2 VGPRs/lane): lanes0-15 hold K=0,1; lanes16-31 K=2,3.
  // A[m][0]=area_row[m], A[m][1]=1, rest 0.
  v2f aop;
  if (lane < 16) { aop.x = rar[lane]; aop.y = 1.0f; }
  else           { aop.x = 0.f;       aop.y = 0.f;  }

  const int half = lane >> 4;
  const int n    = lane & 15;

  unsigned wrow[16];
  #pragma unroll
  for (int r = 0; r < 16; ++r) wrow[r] = 0u;

  #pragma unroll
  for (int h = 0; h < 2; ++h) {
    // B matrix 4x16: B[0][n]=1, B[1][n]=area_col[h*16+n], rest 0.
    v2f bop;
    if (lane < 16) { bop.x = 1.0f; bop.y = car[h * 16 + lane]; }
    else           { bop.x = 0.f;  bop.y = 0.f;                }
    v8f cz = {};
    v8f d = __builtin_amdgcn_wmma_f32_16x16x4_f32(
        false, aop, false, bop, (short)0, cz, false, false);
    // D layout: VGPR r -> M = r + 8*(lane>=16), N = lane&15
    const int col = colBase + h * 16 + n;
    const float cy1 = cbx[h * 16 + n][0], cx1 = cbx[h * 16 + n][1];
    const float cy2 = cbx[h * 16 + n][2], cx2 = cbx[h * 16 + n][3];
    #pragma unroll
    for (int r = 0; r < 8; ++r) {
      const int m = r + 8 * half;
      const int row = rowBase + m;
      float ry1 = rbx[m][0], rx1 = rbx[m][1], ry2 = rbx[m][2], rx2 = rbx[m][3];
      float iy1 = fmaxf(ry1, cy1), ix1 = fmaxf(rx1, cx1);
      float iy2 = fminf(ry2, cy2), ix2 = fminf(rx2, cx2);
      float inter = fmaxf(iy2 - iy1, 0.f) * fmaxf(ix2 - ix1, 0.f);
      float uni = d[r] - inter;                 // area_i + area_j from WMMA
      // division-free IoU threshold test
      bool sup = (inter > NMS_IOU_TH * fmaxf(uni, 1e-8f))
                 && (col > row) && (col < PRE_TOPN);
      unsigned bal = __builtin_amdgcn_ballot_w32(sup);
      // low 16 bits: row r (lanes 0-15); high 16: row r+8 (lanes 16-31)
      wrow[r]     |= (bal & 0xFFFFu) << (16 * h);
      wrow[r + 8] |= (bal >> 16)     << (16 * h);
    }
  }

  if (lane == 0) {
    #pragma unroll
    for (int r = 0; r < 16; ++r) wstage[r] = wrow[r];
  }
  __syncthreads();
  if (lane < 16)
    mask[((size_t)b * KP + rowBase + lane) * MASKW + cb] = wstage[lane];
}

// =====================================================================
// Kernel 4: sequential suppression scan (one block per batch) + output.
// 63-word bitmap in LDS; prefetch next mask row to hide serial latency.
// =====================================================================
__global__ __launch_bounds__(64)
void nms_scan_kernel(const float* __restrict__ selBoxes,
                     const float* __restrict__ selScores,
                     const unsigned* __restrict__ mask,
                     float* __restrict__ outBoxes,   // (B,300,4)
                     float* __restrict__ outScores)  // (B,300)
{
  __shared__ unsigned remv[NBLK];
  __shared__ int keepList[POST_TOPN];
  __shared__ int cnt;
  const int b = blockIdx.x;
  const int tid = threadIdx.x;

  if (tid < NBLK) remv[tid] = 0u;
  if (tid == 0) cnt = 0;
  __syncthreads();

  for (int i = 0; i < PRE_TOPN; ++i) {
    if (tid == 0 && i + 1 < PRE_TOPN)
      __builtin_prefetch(mask + ((size_t)b * KP + i + 1) * MASKW, 0, 1);
    bool alive = ((remv[i >> 5] >> (i & 31)) & 1u) == 0u;   // uniform
    if (alive) {
      if (tid == 0 && cnt < POST_TOPN) { keepList[cnt] = i; cnt = cnt + 1; }
      const unsigned* mr = mask + ((size_t)b * KP + i) * MASKW;
      if (tid < NBLK) remv[tid] |= mr[tid];
    }
    __syncthreads();
  }
  __syncthreads();

  for (int j = tid; j < POST_TOPN; j += 64) {
    float y1 = 0.f, x1 = 0.f, y2 = 0.f, x2 = 0.f, s = 0.f;
    if (j < cnt) {
      int i = keepList[j];
      const float* bp = selBoxes + ((size_t)b * KP + i) * 4;
      y1 = fminf(fmaxf(bp[0], 0.f), 1.f);
      x1 = fminf(fmaxf(bp[1], 0.f), 1.f);
      y2 = fminf(fmaxf(bp[2], 0.f), 1.f);
      x2 = fminf(fmaxf(bp[3], 0.f), 1.f);
      s = selScores[(size_t)b * KP + i];
    }
    float* ob = outBoxes + ((size_t)b * POST_TOPN + j) * 4;
    ob[0] = y1; ob[1] = x1; ob[2] = y2; ob[3] = x2;
    outScores[(size_t)b * POST_TOPN + j] = s;
  }
}

// =====================================================================
extern "C" void kernel_launch(void* const* d_in, const int* in_sizes, int n_in,
                              void* d_out, int out_size, void* d_ws, size_t ws_size,
                              hipStream_t stream) {
  (void)in_sizes; (void)n_in; (void)out_size; (void)ws_size;
  const float* deltas  = (const float*)d_in[0];   // (8,32,256,24)
  const float* logits  = (const float*)d_in[1];   // (8,32,256,6)
  const float* anchors = (const float*)d_in[2];   // (49152,4)

  char* ws = (char*)d_ws;
  size_t off = 0;
  float* scoresAll = (float*)(ws + off); off += (size_t)BATCH * NANCH * sizeof(float);
  float* boxesAll  = (float*)(ws + off); off += (size_t)BATCH * NANCH * 4 * sizeof(float);
  float* selScores = (float*)(ws + off); off += (size_t)BATCH * KP * sizeof(float);
  float* selBoxes  = (float*)(ws + off); off += (size_t)BATCH * KP * 4 * sizeof(float);
  float* selArea   = (float*)(ws + off); off += (size_t)BATCH * KP * sizeof(float);
  unsigned* maskBuf = (unsigned*)(ws + off); off += (size_t)BATCH * KP * MASKW * sizeof(unsigned);

  float* outBoxes  = (float*)d_out;
  float* outScores = outBoxes + (size_t)BATCH * POST_TOPN * 4;

  const int total = BATCH * NANCH;
  decode_softmax_kernel<<<(total + 255) / 256, 256, 0, stream>>>(
      deltas, logits, anchors, scoresAll, boxesAll);
  select_sort_kernel<<<BATCH, 1024, 0, stream>>>(
      scoresAll, boxesAll, selScores, selBoxes, selArea);
  nms_mask_kernel<<<dim3(NBLK, PRE_TOPN / 16, BATCH), 32, 0, stream>>>(
      selBoxes, selArea, maskBuf);
  nms_scan_kernel<<<BATCH, 64, 0, stream>>>(
      selBoxes, selScores, maskBuf, outBoxes, outScores);
}